// TemporalConvBlock_15135464751732
// MI455X (gfx1250) — compile-verified
//
#include <hip/hip_runtime.h>

typedef float v2f __attribute__((ext_vector_type(2)));
typedef float v4f __attribute__((ext_vector_type(4)));
typedef float v8f __attribute__((ext_vector_type(8)));

#define B_    16
#define T_    512
#define F_    64
#define CIN   64
#define COUT  64
#define KW    3
#define DIL   2
#define TM    64            // time rows per block
#define PAD   (DIL*(KW-1))  // 4 (causal left halo)
#define XROWS (TM + PAD)    // 68
#define XSTR  68            // sX row stride (== 4 mod 64 -> conflict-free A frags)
#define WSTR  196           // sW row stride for 192-float rows (== 4 mod 64 -> conflict-free B frags)

#define NW    (COUT * WSTR)          // 12544 floats
#define NSX   (XROWS * XSTR)         // 4624 floats
#define SMEM_FLOATS (NW + NSX + COUT)

// One block = one feature f + 64 time rows of one batch element. 8 waves:
// wave w -> Cout group ng = w&3, M half = w>>2 (two 16-row WMMA tiles sharing
// each B fragment). Reduction order kappa = k*64 + c (tap-major).
// 96 x v_wmma_f32_16x16x4_f32 per wave; all LDS fragment reads conflict-free.
__global__ __launch_bounds__(256)
void tconv_wmma_f32_kernel(const float* __restrict__ x,
                           const float* __restrict__ w,
                           const float* __restrict__ bias,
                           float* __restrict__ y)
{
    extern __shared__ float smem[];
    float* __restrict__ sW = smem;              // raw W[f]: [o][c*3+k], rows padded to WSTR
    float* __restrict__ sX = smem + NW;         // [XROWS][XSTR], cols 0..63 used
    float* __restrict__ sB = smem + NW + NSX;   // [COUT]

    const int f  = blockIdx.x;
    const int bt = blockIdx.y;            // 0..127
    const int b  = bt >> 3;               // T_/TM = 8 tiles per batch element
    const int t0 = (bt & 7) * TM;

    const int tid  = threadIdx.x;
    const int lane = tid & 31;
    const int wave = tid >> 5;

    // --- stage W[f]: coalesced float4 copy into padded rows (lds = flat + 4*o) ---
    const float* wf = w + (size_t)f * (COUT * CIN * KW);
    #pragma unroll
    for (int it = 0; it < (COUT * CIN * KW) / (4 * 256); ++it) {
        int idx = it * 256 + tid;                 // float4 index, 0..3071
        v4f wv = *(const v4f*)(wf + idx * 4);
        int o = idx / 48;                         // 48 float4 per 192-float row
        *(v4f*)(sW + idx * 4 + o * 4) = wv;
    }
    if (tid < COUT) sB[tid] = bias[f * COUT + tid];

    // --- stage x slab (causal zero halo) as float4; coalesced over c ---
    for (int idx = tid; idx < XROWS * (CIN / 4); idx += 256) {
        int r  = idx >> 4;                        // slab row 0..67
        int cq = idx & 15;                        // float4 within row
        int t  = t0 + r - PAD;
        v4f v = { 0.0f, 0.0f, 0.0f, 0.0f };
        if (t >= 0)
            v = *(const v4f*)(x + (((size_t)b * T_ + t) * F_ + f) * CIN + cq * 4);
        *(v4f*)(sX + r * XSTR + cq * 4) = v;
    }
    __syncthreads();

    const int ng = wave & 3;              // Cout group: n0 = ng*16
    const int m0 = (wave >> 2) * 32;      // first of two 16-row M groups
    const int hl = lane >> 4;             // half-wave select
    const int l  = lane & 15;

    float bv = sB[ng * 16 + l];
    v8f acc0 = { bv, bv, bv, bv, bv, bv, bv, bv };
    v8f acc1 = acc0;

    // per-lane W row base: o = ng*16 + l, channel offset 2*hl baked in (x3 = 6)
    const float* wrow = sW + (ng * 16 + l) * WSTR + hl * 6;

    #pragma unroll
    for (int k = 0; k < KW; ++k) {
        // dilated tap rows for the two M groups; channel base 4*c4 + 2*hl
        const float* xr0 = sX + (m0 + l + k * DIL) * XSTR + hl * 2;
        const float* xr1 = xr0 + 16 * XSTR;
        #pragma unroll
        for (int c4 = 0; c4 < CIN / 4; ++c4) {
            v2f a0 = *(const v2f*)(xr0 + c4 * 4);          // ds_load_b64, conflict-free
            v2f a1 = *(const v2f*)(xr1 + c4 * 4);          // ds_load_b64, conflict-free
            v2f bm;
            bm.x = wrow[c4 * 12 + k];                      // W[o][c=4c4+2hl  ][k]
            bm.y = wrow[c4 * 12 + 3 + k];                  // W[o][c=4c4+2hl+1][k]
            acc0 = __builtin_amdgcn_wmma_f32_16x16x4_f32(
                     false, a0, false, bm, (short)0, acc0, false, false);
            acc1 = __builtin_amdgcn_wmma_f32_16x16x4_f32(
                     false, a1, false, bm, (short)0, acc1, false, false);
        }
    }

    // --- store: C/D layout: VGPR r -> M = m0 + r + hl*8 (+16 for acc1), N = ng*16 + l ---
    #pragma unroll
    for (int r = 0; r < 8; ++r) {
        int m = m0 + r + hl * 8;
        size_t row0 = ((size_t)b * T_ + t0 + m) * F_ + f;
        size_t row1 = ((size_t)b * T_ + t0 + m + 16) * F_ + f;
        y[row0 * COUT + ng * 16 + l] = acc0[r];
        y[row1 * COUT + ng * 16 + l] = acc1[r];
    }
}

extern "C" void kernel_launch(void* const* d_in, const int* in_sizes, int n_in,
                              void* d_out, int out_size, void* d_ws, size_t ws_size,
                              hipStream_t stream) {
    const float* x    = (const float*)d_in[0];  // [B,T,F,Cin]
    const float* w    = (const float*)d_in[1];  // [F,Cout,Cin,K]
    const float* bias = (const float*)d_in[2];  // [F,Cout]
    float*       y    = (float*)d_out;          // [B,T,F,Cout]

    dim3 grid(F_, (B_ * T_) / TM, 1);
    size_t shmem = SMEM_FLOATS * sizeof(float); // ~67.3 KB dynamic LDS (gfx1250: 320 KB/WGP)
    tconv_wmma_f32_kernel<<<grid, 256, shmem, stream>>>(x, w, bias, y);
}